// NG_29652454212096
// MI455X (gfx1250) — compile-verified
//
#include <hip/hip_runtime.h>
#include <hip/hip_bf16.h>
#include <stdint.h>

// out[h,w] = W[15,0] * x[h,w]  (reference einsum collapses: state is zero
// except state[0]=x, so next_state[15] = W[15,0]*x). Pure streaming scale:
// 64 MiB of traffic -> ~2.9us at 23.3 TB/s. We drive it through the CDNA5
// Tensor Data Mover: per-wave double-buffered tensor_load_to_lds pipeline,
// split-wait on TENSORcnt, ds_load_b128 consume, global_store_b128 produce.

typedef unsigned int v4u __attribute__((ext_vector_type(4)));
typedef int          v8i __attribute__((ext_vector_type(8)));
typedef int          v4i __attribute__((ext_vector_type(4)));

#define TILE_F4         256                  // float4 elements per tile
#define TILE_FLOATS     (TILE_F4 * 4)        // 1024 floats = 4 KiB per TDM op
#define TILE_BYTES      (TILE_FLOATS * 4)
#define WAVES_PER_BLOCK 8
#define BLOCK_THREADS   (WAVES_PER_BLOCK * 32)
#define GRID_BLOCKS     512

// Build a 1-D Tensor DMA Descriptor (D#) for one tile and issue
// TENSOR_LOAD_TO_LDS. All inputs are wave-uniform -> descriptor lands in
// SGPRs. Layout per CDNA5 ISA ch.8 (groups 0..3).
__device__ __forceinline__ void tdm_load_tile(const float* gsrc, uint32_t lds_byte_off) {
    uint64_t ga = (uint64_t)(uintptr_t)gsrc;

    v4u g0;
    g0[0] = 1u;                                               // count=1 (valid), user mode
    g0[1] = lds_byte_off;                                     // lds_addr (bytes)
    g0[2] = (uint32_t)ga;                                     // global_addr[31:0]
    g0[3] = (uint32_t)((ga >> 32) & 0x1FFFFFFu) | (2u << 30); // global_addr[56:32] | type=2

    v8i g1;
    g1[0] = (int)(2u << 16);                                  // data_size=2 (4 bytes); no multicast/pad/iterate
    g1[1] = (int)((TILE_FLOATS & 0xFFFFu) << 16);             // tensor_dim0[15:0] (tile==tensor: no OOB)
    g1[2] = (int)(((TILE_FLOATS >> 16) & 0xFFFFu) | (1u << 16)); // tensor_dim0[31:16] | tensor_dim1=1
    g1[3] = (int)((TILE_FLOATS & 0xFFFFu) << 16);             // tensor_dim1 hi (0) | tile_dim0=1024
    g1[4] = 0;                                                // tile_dim1=0 (unused), tile_dim2=0 (unused)
    g1[5] = (int)TILE_FLOATS;                                 // tensor_dim0_stride[31:0]
    g1[6] = 0;                                                // stride0 hi | stride1 lo
    g1[7] = 0;                                                // stride1 hi

    v4i g2;
    g2[0] = 1;                // tensor_dim2 = 1 (avoid spurious OOB in unused dims)
    g2[1] = 1;                // tensor_dim3 = 1
    g2[2] = 0;                // tensor_dim2_stride lo
    g2[3] = 0;                // stride hi | tile_dim3=0 (unused)

    v4i g3;
    g3[0] = 0;                // tensor_dim3_stride lo
    g3[1] = (int)(1u << 16);  // stride hi | tensor_dim4 = 1
    g3[2] = 0;                // tensor_dim4 hi | tile_dim4=0 (unused)
    g3[3] = 0;

    // Trailing group required by the 6-arg (clang-23 / therock-10.0) builtin
    // signature; zero-filled for a plain non-gather, non-iterating load.
    v8i g4;
    g4[0] = 0; g4[1] = 0; g4[2] = 0; g4[3] = 0;
    g4[4] = 0; g4[5] = 0; g4[6] = 0; g4[7] = 0;

    __builtin_amdgcn_tensor_load_to_lds(g0, g1, g2, g3, g4, /*cpol=*/0);
}

__global__ __launch_bounds__(BLOCK_THREADS)
void ng_scale_tdm_kernel(const float* __restrict__ x,
                         const float* __restrict__ Wm,
                         float* __restrict__ out,
                         int numTiles, int n) {
    // Wave-uniform scalar: W[15,0] -> SMEM load, broadcast.
    const float s = Wm[15 * 16 + 0];

    const int lane = threadIdx.x & 31;
    const int wave = (int)(threadIdx.x >> 5);

    // Per-wave double buffer: [wave][stage][TILE_F4]
    __shared__ float4 lds_tiles[WAVES_PER_BLOCK * 2 * TILE_F4];
    const uint32_t wave_base =
        (uint32_t)(uintptr_t)(&lds_tiles[wave * 2 * TILE_F4]);

    const int gw      = (int)blockIdx.x * WAVES_PER_BLOCK + wave;  // global wave id
    const int wstride = (int)gridDim.x * WAVES_PER_BLOCK;

    int stage = 0;
    int t = gw;
    if (t < numTiles) {
        tdm_load_tile(x + (size_t)t * TILE_FLOATS, wave_base);
    }

    for (; t < numTiles; t += wstride) {
        const int nxt = t + wstride;
        if (nxt < numTiles) {
            // Prefetch next tile into the other stage, then wait until only
            // that one is outstanding (tensor ops complete in-order per wave
            // => current tile's data is in LDS).
            tdm_load_tile(x + (size_t)nxt * TILE_FLOATS,
                          wave_base + (uint32_t)((stage ^ 1) * TILE_BYTES));
            __builtin_amdgcn_s_wait_tensorcnt(1);
        } else {
            __builtin_amdgcn_s_wait_tensorcnt(0);
        }

        const float4* src = &lds_tiles[(wave * 2 + stage) * TILE_F4];
        float4* dst = (float4*)out + (size_t)t * TILE_F4;

#pragma unroll
        for (int k = 0; k < TILE_F4 / 32; ++k) {
            float4 v = src[k * 32 + lane];     // ds_load_b128, conflict-free
            v.x *= s; v.y *= s; v.z *= s; v.w *= s;
            dst[k * 32 + lane] = v;            // global_store_b128, coalesced
        }
        stage ^= 1;
    }

    // Defensive scalar tail (n is divisible by TILE_FLOATS for this problem).
    const int tail = numTiles * TILE_FLOATS;
    for (int i = tail + (int)(blockIdx.x * blockDim.x + threadIdx.x); i < n;
         i += (int)(gridDim.x * blockDim.x)) {
        out[i] = s * x[i];
    }
}

extern "C" void kernel_launch(void* const* d_in, const int* in_sizes, int n_in,
                              void* d_out, int out_size, void* d_ws, size_t ws_size,
                              hipStream_t stream) {
    const float* x  = (const float*)d_in[0];   // [2048, 4096] fp32
    const float* Wm = (const float*)d_in[1];   // [16, 16] fp32
    float* out      = (float*)d_out;           // [2048, 4096] fp32

    const int n        = in_sizes[0];
    const int numTiles = n / TILE_FLOATS;

    ng_scale_tdm_kernel<<<dim3(GRID_BLOCKS), dim3(BLOCK_THREADS), 0, stream>>>(
        x, Wm, out, numTiles, n);
}